// GATEConv_with_global_pool_19121194402157
// MI455X (gfx1250) — compile-verified
//
#include <hip/hip_runtime.h>
#include <hip/hip_bf16.h>

// ---------------------------------------------------------------------------
// GATEConv GNN (3 layers) + interface pooling + MLP for gfx1250 (CDNA5).
// Dense node transforms use V_WMMA_F32_16X16X4_F32 (fp32-exact matrix pipe).
// Edge softmax/message passing via atomics (L2-resident accumulators).
// ---------------------------------------------------------------------------

#define NN     50000
#define F_IN   32
#define HID    64
#define HC     128      // HEADS * HID
#define NEA    2
#define EE     800000
#define ET     (EE + NN)   // with self loops
#define NG     50
#define NI_PER 200         // interface nodes per graph
#define SLOPE  0.2f

typedef __attribute__((ext_vector_type(2))) float v2f;
typedef __attribute__((ext_vector_type(8))) float v8f;

// ---------------- float atomic max via monotone int mapping ----------------
__device__ inline void atomicMaxFloat(float* addr, float val) {
  if (val >= 0.0f) {
    atomicMax((int*)addr, __float_as_int(val));
  } else {
    atomicMin((unsigned int*)addr, __float_as_uint(val));
  }
}

__device__ inline float leaky(float v) { return v > 0.0f ? v : SLOPE * v; }

// ---------------------------------------------------------------------------
// GEMM: C[N x 128] = A[N x K] * B[K x 128], fp32 WMMA 16x16x4.
// One wave per 16x16 output tile; block = 256 threads = 8 waves = 8 col tiles,
// grid.x = N/16 row tiles. N % 16 == 0 so EXEC is always all-ones for WMMA.
// ---------------------------------------------------------------------------
template <int K>
__global__ void gemm_wmma_f32(const float* __restrict__ A,
                              const float* __restrict__ B,
                              float* __restrict__ C) {
  const int lane  = threadIdx.x & 31;
  const int ntile = threadIdx.x >> 5;   // 0..7 -> col tile
  const int mtile = blockIdx.x;         // row tile
  const int m0 = mtile * 16;
  const int n0 = ntile * 16;
  const int half = lane >> 4;           // 0: lanes 0-15, 1: lanes 16-31
  const int l16  = lane & 15;

  const float* arow = A + (size_t)(m0 + l16) * K;
  v8f c = {};
#pragma unroll
  for (int k0 = 0; k0 < K; k0 += 4) {
    const int ka = k0 + 2 * half;       // {0,1} or {2,3} within the 4-K step
    v2f a, b;
    a.x = arow[ka];
    a.y = arow[ka + 1];
    b.x = B[(size_t)ka * HC + n0 + l16];
    b.y = B[(size_t)(ka + 1) * HC + n0 + l16];
    c = __builtin_amdgcn_wmma_f32_16x16x4_f32(false, a, false, b,
                                              (short)0, c, false, false);
  }
  float* crow = C + n0 + l16;
#pragma unroll
  for (int i = 0; i < 8; ++i) {
    crow[(size_t)(m0 + i + 8 * half) * HC] = c[i];
  }
}

// ---------------------------------------------------------------------------
// al[n][h] = sum_c xl[n][h*64+c]*att_l[h][c]; same for ar. Thread per (n,h).
// ---------------------------------------------------------------------------
__global__ void node_attn_kernel(const float* __restrict__ xl,
                                 const float* __restrict__ attl,
                                 const float* __restrict__ attr,
                                 float* __restrict__ al,
                                 float* __restrict__ ar) {
  int i = blockIdx.x * blockDim.x + threadIdx.x;
  if (i >= NN * 2) return;
  int n = i >> 1, h = i & 1;
  const float* xr = xl + (size_t)n * HC + h * HID;
  const float* wl = attl + h * HID;
  const float* wr = attr + h * HID;
  float sl = 0.0f, sr = 0.0f;
#pragma unroll 8
  for (int c = 0; c < HID; ++c) {
    float x = xr[c];
    sl += x * wl[c];
    sr += x * wr[c];
  }
  al[i] = sl;
  ar[i] = sr;
}

// init per-layer accumulators: out=0 (float4 stores), m=-inf, d=0
__global__ void init_accum_kernel(float4* __restrict__ outacc4,
                                  float* __restrict__ m,
                                  float* __restrict__ d) {
  int i = blockIdx.x * blockDim.x + threadIdx.x;
  if (i < NN * (HC / 4)) outacc4[i] = make_float4(0.f, 0.f, 0.f, 0.f);
  if (i < NN * 2) { m[i] = -__builtin_inff(); d[i] = 0.0f; }
}

// ---------------------------------------------------------------------------
// softmax pass A: per-dst max of leaky(al[src]+ar[dst]) (atomic float max).
// ---------------------------------------------------------------------------
__global__ void edge_max_kernel(const int* __restrict__ src,
                                const int* __restrict__ dst,
                                const float* __restrict__ al,
                                const float* __restrict__ ar,
                                float* __restrict__ m) {
  int e = blockIdx.x * blockDim.x + threadIdx.x;
  if (e >= ET) return;
  int s, t;
  if (e < EE) { s = src[e]; t = dst[e]; } else { s = t = e - EE; }
#pragma unroll
  for (int h = 0; h < 2; ++h) {
    float v = leaky(al[s * 2 + h] + ar[t * 2 + h]);
    atomicMaxFloat(&m[t * 2 + h], v);
  }
}

// softmax pass B: per-dst sum of exp(e - m[dst]).
__global__ void edge_expsum_kernel(const int* __restrict__ src,
                                   const int* __restrict__ dst,
                                   const float* __restrict__ al,
                                   const float* __restrict__ ar,
                                   const float* __restrict__ m,
                                   float* __restrict__ d) {
  int e = blockIdx.x * blockDim.x + threadIdx.x;
  if (e >= ET) return;
  int s, t;
  if (e < EE) { s = src[e]; t = dst[e]; } else { s = t = e - EE; }
#pragma unroll
  for (int h = 0; h < 2; ++h) {
    float v = leaky(al[s * 2 + h] + ar[t * 2 + h]);
    atomicAdd(&d[t * 2 + h], expf(v - m[t * 2 + h]));
  }
}

// ---------------------------------------------------------------------------
// message pass: one wave per edge; each lane owns 4 CONTIGUOUS channels
// (j = lane*4..lane*4+3) so the xl[src]/eW/eb gathers are single b128 loads
// with full wave coalescing. msg = (xl[src] + ea@eW + eb) * alpha,
// scatter-added to out[dst] (per-dword f32 atomics, L2 resident).
// ---------------------------------------------------------------------------
__global__ void edge_message_kernel(const int* __restrict__ src,
                                    const int* __restrict__ dst,
                                    const float* __restrict__ eattr,
                                    const float* __restrict__ al,
                                    const float* __restrict__ ar,
                                    const float* __restrict__ m,
                                    const float* __restrict__ d,
                                    const float* __restrict__ xl,
                                    const float* __restrict__ eW,  // [2,128]
                                    const float* __restrict__ eb,  // [128]
                                    float* __restrict__ outacc) {
  int gid  = blockIdx.x * blockDim.x + threadIdx.x;
  int e    = gid >> 5;
  int lane = gid & 31;
  if (e >= ET) return;
  int s, t;
  float ea0, ea1;
  if (e < EE) {
    s = src[e]; t = dst[e];
    ea0 = 1.0f / eattr[(size_t)e * 2];
    ea1 = 1.0f / eattr[(size_t)e * 2 + 1];
  } else {
    s = t = e - EE;
    ea0 = ea1 = 0.0f;
  }
  // per-half-wave (per-head) scalar part (one edge per wave -> loads coalesce)
  int j0 = lane * 4;                 // 4 contiguous channels per lane
  float alpha = 0.0f;
  {
    int h = j0 >> 6;                 // head for this lane's channel group
    float v = leaky(al[s * 2 + h] + ar[t * 2 + h]);
    alpha = expf(v - m[t * 2 + h]) / (d[t * 2 + h] + 1e-16f);
  }
  const float4 xs  = *(const float4*)(xl + (size_t)s * HC + j0);
  const float4 w0  = *(const float4*)(eW + j0);
  const float4 w1  = *(const float4*)(eW + HC + j0);
  const float4 bb  = *(const float4*)(eb + j0);
  float* od = outacc + (size_t)t * HC + j0;
  atomicAdd(&od[0], (xs.x + ea0 * w0.x + ea1 * w1.x + bb.x) * alpha);
  atomicAdd(&od[1], (xs.y + ea0 * w0.y + ea1 * w1.y + bb.y) * alpha);
  atomicAdd(&od[2], (xs.z + ea0 * w0.z + ea1 * w1.z + bb.z) * alpha);
  atomicAdd(&od[3], (xs.w + ea0 * w0.w + ea1 * w1.w + bb.w) * alpha);
}

// head-mean + bias + tanh; also JumpingKnowledge max.
__global__ void combine_kernel(const float* __restrict__ outacc,
                               const float* __restrict__ bconv, // [64]
                               float* __restrict__ h,
                               float* __restrict__ xmax,
                               int first) {
  int i = blockIdx.x * blockDim.x + threadIdx.x;
  if (i >= NN * HID) return;
  int n = i >> 6, c = i & 63;
  float v = tanhf(0.5f * (outacc[(size_t)n * HC + c] +
                          outacc[(size_t)n * HC + HID + c]) + bconv[c]);
  h[i] = v;
  xmax[i] = first ? v : fmaxf(xmax[i], v);
}

// ---------------------------------------------------------------------------
// Per-graph pooling: add / mean / attention / max over 200 interface nodes.
// One block (256 threads) per graph.
// ---------------------------------------------------------------------------
__global__ void pool_kernel(const float* __restrict__ xmax,
                            const int* __restrict__ ipos,
                            const float* __restrict__ gate_w, // [64]
                            const float* __restrict__ gate_b, // [1]
                            float* __restrict__ pooled) {     // [G,256]
  __shared__ float gate[NI_PER];
  __shared__ float red[256];
  int g = blockIdx.x;
  int t = threadIdx.x;

  if (t < NI_PER) {
    int node = ipos[g * NI_PER + t];
    const float* xr = xmax + (size_t)node * HID;
    float sdot = gate_b[0];
#pragma unroll 8
    for (int c = 0; c < HID; ++c) sdot += xr[c] * gate_w[c];
    gate[t] = sdot;
  }
  __syncthreads();

  red[t] = (t < NI_PER) ? gate[t] : -__builtin_inff();
  __syncthreads();
  for (int off = 128; off > 0; off >>= 1) {
    if (t < off) red[t] = fmaxf(red[t], red[t + off]);
    __syncthreads();
  }
  float gmax = red[0];
  __syncthreads();

  float ex = (t < NI_PER) ? expf(gate[t] - gmax) : 0.0f;
  red[t] = ex;
  __syncthreads();
  for (int off = 128; off > 0; off >>= 1) {
    if (t < off) red[t] += red[t + off];
    __syncthreads();
  }
  float denom = red[0];
  __syncthreads();
  if (t < NI_PER) gate[t] = ex / (denom + 1e-16f);
  __syncthreads();

  if (t < HID) {
    float add = 0.0f, attp = 0.0f, mx = -__builtin_inff();
    for (int k = 0; k < NI_PER; ++k) {
      int node = ipos[g * NI_PER + k];
      float x = xmax[(size_t)node * HID + t];
      add += x;
      attp += gate[k] * x;
      mx = fmaxf(mx, x);
    }
    pooled[g * 256 + t]           = add;
    pooled[g * 256 + HID + t]     = add * (1.0f / NI_PER);
    pooled[g * 256 + 2 * HID + t] = attp;
    pooled[g * 256 + 3 * HID + t] = mx;
  }
}

// MLP head: out[g] = (tanh(pooled @ W1 + b1)) @ W2 + b2. One 128-thr block/graph.
__global__ void mlp_kernel(const float* __restrict__ pooled,
                           const float* __restrict__ w1, // [256,128]
                           const float* __restrict__ b1, // [128]
                           const float* __restrict__ w2, // [128]
                           const float* __restrict__ b2, // [1]
                           float* __restrict__ out) {
  __shared__ float p[256];
  __shared__ float red[128];
  int g = blockIdx.x;
  int t = threadIdx.x;
  p[t]       = pooled[g * 256 + t];
  p[t + 128] = pooled[g * 256 + 128 + t];
  __syncthreads();
  float s = b1[t];
  for (int i = 0; i < 256; ++i) s += p[i] * w1[i * 128 + t];
  red[t] = tanhf(s) * w2[t];
  __syncthreads();
  for (int off = 64; off > 0; off >>= 1) {
    if (t < off) red[t] += red[t + off];
    __syncthreads();
  }
  if (t == 0) out[g] = red[0] + b2[0];
}

// ---------------------------------------------------------------------------
extern "C" void kernel_launch(void* const* d_in, const int* in_sizes, int n_in,
                              void* d_out, int out_size, void* d_ws, size_t ws_size,
                              hipStream_t stream) {
  (void)in_sizes; (void)n_in; (void)out_size; (void)ws_size;

  const float* x      = (const float*)d_in[0];
  const int*   eidx   = (const int*)d_in[1];     // [2,E]
  const float* eattr  = (const float*)d_in[2];   // [E,2]
  // d_in[3] = batch (derived -> unused), d_in[5] = graph_num (constant 50)
  const int*   ipos   = (const int*)d_in[4];     // [10000]
  const float* W0     = (const float*)d_in[6];   // [32,128]
  const float* attl0  = (const float*)d_in[7];   // [2,64]
  const float* attr0  = (const float*)d_in[8];
  const float* W12    = (const float*)d_in[9];   // [2,64,128]
  const float* attl12 = (const float*)d_in[10];  // [2,2,64]
  const float* attr12 = (const float*)d_in[11];
  const float* eW     = (const float*)d_in[12];  // [3,2,128]
  const float* eb     = (const float*)d_in[13];  // [3,128]
  const float* bconv  = (const float*)d_in[14];  // [3,64]
  const float* gate_w = (const float*)d_in[15];  // [64,1]
  const float* gate_b = (const float*)d_in[16];  // [1]
  const float* lin1_w = (const float*)d_in[17];  // [256,128]
  const float* lin1_b = (const float*)d_in[18];  // [128]
  const float* lin2_w = (const float*)d_in[19];  // [128,1]
  const float* lin2_b = (const float*)d_in[20];  // [1]
  float* out = (float*)d_out;

  const int* src = eidx;
  const int* dst = eidx + EE;

  // workspace carving (floats)
  float* ws     = (float*)d_ws;
  float* xl     = ws;                       // N*128
  float* outacc = xl + (size_t)NN * HC;     // N*128
  float* hbuf   = outacc + (size_t)NN * HC; // N*64
  float* xmax   = hbuf + (size_t)NN * HID;  // N*64
  float* alv    = xmax + (size_t)NN * HID;  // N*2
  float* arv    = alv + NN * 2;             // N*2
  float* mmax   = arv + NN * 2;             // N*2
  float* dsum   = mmax + NN * 2;            // N*2
  float* pooled = dsum + NN * 2;            // G*256

  const int B = 256;
  const int gInit  = (NN * (HC / 4) + B - 1) / B;
  const int gAttn  = (NN * 2 + B - 1) / B;
  const int gEdge  = (ET + B - 1) / B;
  const int gMsg   = (int)(((size_t)ET * 32 + B - 1) / B);
  const int gComb  = (NN * HID + B - 1) / B;

  for (int l = 0; l < 3; ++l) {
    // 1) xl = h @ W  (fp32 WMMA)
    if (l == 0) {
      gemm_wmma_f32<F_IN><<<NN / 16, B, 0, stream>>>(x, W0, xl);
    } else {
      gemm_wmma_f32<HID><<<NN / 16, B, 0, stream>>>(
          hbuf, W12 + (size_t)(l - 1) * HID * HC, xl);
    }
    // 2) attention logits per node
    const float* al_w = (l == 0) ? attl0 : attl12 + (size_t)(l - 1) * 2 * HID;
    const float* ar_w = (l == 0) ? attr0 : attr12 + (size_t)(l - 1) * 2 * HID;
    node_attn_kernel<<<gAttn, B, 0, stream>>>(xl, al_w, ar_w, alv, arv);
    // 3) reset accumulators
    init_accum_kernel<<<gInit, B, 0, stream>>>((float4*)outacc, mmax, dsum);
    // 4) segment softmax (max, then exp-sum)
    edge_max_kernel<<<gEdge, B, 0, stream>>>(src, dst, alv, arv, mmax);
    edge_expsum_kernel<<<gEdge, B, 0, stream>>>(src, dst, alv, arv, mmax, dsum);
    // 5) weighted message scatter (coalesced b128 gathers)
    edge_message_kernel<<<gMsg, B, 0, stream>>>(
        src, dst, eattr, alv, arv, mmax, dsum, xl,
        eW + (size_t)l * 2 * HC, eb + (size_t)l * HC, outacc);
    // 6) head mean + bias + tanh, JK max
    combine_kernel<<<gComb, B, 0, stream>>>(outacc, bconv + (size_t)l * HID,
                                            hbuf, xmax, l == 0 ? 1 : 0);
  }

  pool_kernel<<<NG, 256, 0, stream>>>(xmax, ipos, gate_w, gate_b, pooled);
  mlp_kernel<<<NG, 128, 0, stream>>>(pooled, lin1_w, lin1_b, lin2_w, lin2_b, out);
}